// ScaledDotProductAttention_5738076307480
// MI455X (gfx1250) — compile-verified
//
#include <hip/hip_runtime.h>
#include <math.h>

#define BATCH 2
#define HEADS 16
#define SEQ   2048
#define DIM   128
#define BH    (BATCH * HEADS)
#define T16   (SEQ / 16)                 // 128 tiles of 16 along S
#define INV_SCALE 0.08838834764831845f   // 1/sqrt(128)

#define KSTR 132   // K/Q tile LDS row stride (words): b64 frag reads conflict-free
#define VSTR 136   // V tile LDS row stride (words): b32 frag reads conflict-free
#define PSTR 18    // P tile stride: 8B-aligned b64, conflict-free

#define KROW_B (KSTR * 4)   // 528 bytes, 16B aligned
#define VROW_B (VSTR * 4)   // 544 bytes, 16B aligned
#define KTILE_B (16 * KROW_B)
#define VTILE_B (16 * VROW_B)

typedef float v2f __attribute__((ext_vector_type(2)));
typedef float v8f __attribute__((ext_vector_type(8)));
#define Z8 ((v8f){0.f,0.f,0.f,0.f,0.f,0.f,0.f,0.f})

static __device__ __forceinline__ v2f ldg2(const float* p) {
  return *reinterpret_cast<const v2f*>(p);
}

// ---- CDNA5 async global->LDS copy (16B/lane), tracked by ASYNCcnt ----------
static __device__ __forceinline__ void async_b128(unsigned lds_byte_addr,
                                                  unsigned voff_bytes,
                                                  const float* sbase) {
  asm volatile("global_load_async_to_lds_b128 %0, %1, %2"
               :: "v"(lds_byte_addr), "v"(voff_bytes), "s"(sbase)
               : "memory");
}
static __device__ __forceinline__ void wait_async0() {
  asm volatile("s_wait_asynccnt 0x0" ::: "memory");
}

// ---------------------------------------------------------------------------
// Pass 1: per-column stats of S = Q K^T / sqrt(d):  m_k and 1/D_k.
// Wave owns 16 k-columns (K fragments pinned in 64 VGPRs); the block streams
// the SAME q-tiles, staged cooperatively via async-to-LDS, double buffered.
// ---------------------------------------------------------------------------
__global__ __launch_bounds__(256) void colsoftmax_stats_kernel(
    const float* __restrict__ Q, const float* __restrict__ K,
    float* __restrict__ Mout, float* __restrict__ Dinv) {
  __shared__ __attribute__((aligned(16))) float qlds[2][16 * KSTR];

  const int tid  = threadIdx.x;
  const int wave = tid >> 5;
  const int lane = tid & 31;
  const int n    = lane & 15;
  const int hl   = lane >> 4;
  const int gw   = blockIdx.x * 8 + wave;
  const int bh   = gw / T16;
  const int kt   = gw % T16;
  const int kb   = kt * 16;

  const float* Qh = Q + (size_t)bh * SEQ * DIM;
  const float* Kh = K + (size_t)bh * SEQ * DIM;

  // pinned B operand: lane n = column k = kb+n, K-slots 2*hl,2*hl+1 per chunk
  v2f bk[32];
  {
    const float* kp = Kh + (size_t)(kb + n) * DIM + 2 * hl;
#pragma unroll
    for (int c = 0; c < 32; ++c) bk[c] = ldg2(kp + 4 * c);
  }

  // this thread's two 16B transfers per 16x128 tile (512 transfers total)
  const unsigned r0 = (unsigned)tid >> 5, c4 = (unsigned)tid & 31, r1 = r0 + 8;
  const unsigned voff0 = r0 * (DIM * 4) + c4 * 16;
  const unsigned voff1 = r1 * (DIM * 4) + c4 * 16;
  const unsigned l0 = r0 * KROW_B + c4 * 16;
  const unsigned l1 = r1 * KROW_B + c4 * 16;
  const unsigned qbase = (unsigned)(uintptr_t)&qlds[0][0];

  // prologue: stage q-tile 0 into buffer 0
  async_b128(qbase + l0, voff0, Qh);
  async_b128(qbase + l1, voff1, Qh);

  float mx = -INFINITY, dn = 0.0f;

  for (int qt = 0; qt < T16; ++qt) {
    const int cur = qt & 1;
    wait_async0();        // my transfers for buffer `cur` have landed
    __syncthreads();      // everyone's have; previous compute on cur^1 is done
    if (qt + 1 < T16) {   // prefetch next tile into the other buffer
      const float* nb = Qh + (size_t)(qt + 1) * 16 * DIM;
      const unsigned b = qbase + (unsigned)(cur ^ 1) * KTILE_B;
      async_b128(b + l0, voff0, nb);
      async_b128(b + l1, voff1, nb);
    }

    // S^tile (16q x 16k): A from LDS, B pinned; 4 rotating accumulators
    const float* qr = &qlds[cur][n * KSTR + 2 * hl];
    v8f acc[4] = {Z8, Z8, Z8, Z8};
#pragma unroll
    for (int c = 0; c < 32; ++c) {
      v2f a = *reinterpret_cast<const v2f*>(qr + 4 * c);
      acc[c & 3] = __builtin_amdgcn_wmma_f32_16x16x4_f32(
          false, a, false, bk[c], (short)0, acc[c & 3], false, false);
    }

    // online max / sum(exp) along q for this lane's column
    float s[8];
    float tm = -INFINITY;
#pragma unroll
    for (int r = 0; r < 8; ++r) {
      s[r] = ((acc[0][r] + acc[1][r]) + (acc[2][r] + acc[3][r])) * INV_SCALE;
      tm = fmaxf(tm, s[r]);
    }
    const float nm = fmaxf(mx, tm);
    float sum = 0.0f;
#pragma unroll
    for (int r = 0; r < 8; ++r) sum += __expf(s[r] - nm);
    dn = dn * __expf(mx - nm) + sum;
    mx = nm;
  }

  // combine half-wave partials (lane <-> lane^16 share the same column)
  const float mo = __shfl_xor(mx, 16, 32);
  const float dd = __shfl_xor(dn, 16, 32);
  const float mt = fmaxf(mx, mo);
  const float dt = dn * __expf(mx - mt) + dd * __expf(mo - mt);
  if (hl == 0) {
    Mout[(size_t)bh * SEQ + kb + n] = mt;
    Dinv[(size_t)bh * SEQ + kb + n] = 1.0f / dt;
  }
}

// ---------------------------------------------------------------------------
// Pass 2: O = P V, P[q,k] = exp(S[q,k]-m_k)/D_k.
// Wave owns 16(q) x 128(d); Q fragments pinned in 64 VGPRs. K and V tiles are
// identical for all 8 waves -> staged once per block via async-to-LDS (double
// buffered). P goes C-layout -> A-layout through a wave-private padded tile.
// ---------------------------------------------------------------------------
__global__ __launch_bounds__(256) void colsoftmax_av_kernel(
    const float* __restrict__ Q, const float* __restrict__ K,
    const float* __restrict__ V,
    const float* __restrict__ Mw, const float* __restrict__ Dinv,
    float* __restrict__ Out) {
  __shared__ __attribute__((aligned(16))) float klds[2][16 * KSTR];
  __shared__ __attribute__((aligned(16))) float vlds[2][16 * VSTR];
  __shared__ __attribute__((aligned(16))) float plds[8][16 * PSTR];

  const int tid  = threadIdx.x;
  const int wave = tid >> 5;
  const int lane = tid & 31;
  const int n    = lane & 15;
  const int hl   = lane >> 4;
  const int gw   = blockIdx.x * 8 + wave;
  const int bh   = gw / T16;
  const int qt   = gw % T16;
  const int qb   = qt * 16;

  const float* Qh = Q + (size_t)bh * SEQ * DIM;
  const float* Kh = K + (size_t)bh * SEQ * DIM;
  const float* Vh = V + (size_t)bh * SEQ * DIM;

  // pinned A operand: lane n = row q = qb+n
  v2f qa[32];
  {
    const float* qp = Qh + (size_t)(qb + n) * DIM + 2 * hl;
#pragma unroll
    for (int c = 0; c < 32; ++c) qa[c] = ldg2(qp + 4 * c);
  }

  v8f oacc[8];
#pragma unroll
  for (int d = 0; d < 8; ++d) oacc[d] = Z8;

  // async transfer geometry (512 x 16B per 16x128 tile)
  const unsigned r0 = (unsigned)tid >> 5, c4t = (unsigned)tid & 31, r1 = r0 + 8;
  const unsigned voff0 = r0 * (DIM * 4) + c4t * 16;
  const unsigned voff1 = r1 * (DIM * 4) + c4t * 16;
  const unsigned kl0 = r0 * KROW_B + c4t * 16, kl1 = r1 * KROW_B + c4t * 16;
  const unsigned vl0 = r0 * VROW_B + c4t * 16, vl1 = r1 * VROW_B + c4t * 16;
  const unsigned kbase = (unsigned)(uintptr_t)&klds[0][0];
  const unsigned vbase = (unsigned)(uintptr_t)&vlds[0][0];

  // prologue: stage k-tile 0
  async_b128(kbase + kl0, voff0, Kh);
  async_b128(kbase + kl1, voff1, Kh);
  async_b128(vbase + vl0, voff0, Vh);
  async_b128(vbase + vl1, voff1, Vh);

  float* pw = plds[wave];

  for (int kt = 0; kt < T16; ++kt) {
    const int cur = kt & 1;
    const int kb  = kt * 16;
    wait_async0();
    __syncthreads();
    if (kt + 1 < T16) {
      const float* nk = Kh + (size_t)(kt + 1) * 16 * DIM;
      const float* nv = Vh + (size_t)(kt + 1) * 16 * DIM;
      const unsigned kbuf = kbase + (unsigned)(cur ^ 1) * KTILE_B;
      const unsigned vbuf = vbase + (unsigned)(cur ^ 1) * VTILE_B;
      async_b128(kbuf + kl0, voff0, nk);
      async_b128(kbuf + kl1, voff1, nk);
      async_b128(vbuf + vl0, voff0, nv);
      async_b128(vbuf + vl1, voff1, nv);
    }

    // S tile = Q_tile . K_tile^T  (B fragments from LDS, broadcast to all waves)
    const float* kr = &klds[cur][n * KSTR + 2 * hl];
    v8f acc[4] = {Z8, Z8, Z8, Z8};
#pragma unroll
    for (int c = 0; c < 32; ++c) {
      v2f b = *reinterpret_cast<const v2f*>(kr + 4 * c);
      acc[c & 3] = __builtin_amdgcn_wmma_f32_16x16x4_f32(
          false, qa[c], false, b, (short)0, acc[c & 3], false, false);
    }

    // lane's column k is fixed -> stats are per-lane scalars
    const float mn = Mw  [(size_t)bh * SEQ + kb + n];
    const float di = Dinv[(size_t)bh * SEQ + kb + n];

    // P = exp(S/scale - m_k) / D_k ; spill C-layout -> LDS row-major [q][k]
#pragma unroll
    for (int r = 0; r < 8; ++r) {
      const float sv = (acc[0][r] + acc[1][r]) + (acc[2][r] + acc[3][r]);
      pw[(r + 8 * hl) * PSTR + n] = __expf(sv * INV_SCALE - mn) * di;
    }
    __builtin_amdgcn_wave_barrier();   // wave-private region; DS in-order per wave

    // reload P in A-layout: lane n = row q, K-slots 4*c4 + 2*hl + {0,1}
    v2f pa[4];
#pragma unroll
    for (int c4 = 0; c4 < 4; ++c4)
      pa[c4] = *reinterpret_cast<const v2f*>(&pw[n * PSTR + 4 * c4 + 2 * hl]);

    // O += P . V  (B fragments from staged V tile; 8 independent chains)
    const float* vr = &vlds[cur][(2 * hl) * VSTR + n];
#pragma unroll
    for (int ds = 0; ds < 8; ++ds) {
#pragma unroll
      for (int c4 = 0; c4 < 4; ++c4) {
        v2f bv;
        bv.x = vr[(4 * c4)     * VSTR + ds * 16];
        bv.y = vr[(4 * c4 + 1) * VSTR + ds * 16];
        oacc[ds] = __builtin_amdgcn_wmma_f32_16x16x4_f32(
            false, pa[c4], false, bv, (short)0, oacc[ds], false, false);
      }
    }
  }

  // write O from C-layout: lane column d = ds*16+n, rows q = r + 8*hl
  float* op = Out + (size_t)bh * SEQ * DIM;
#pragma unroll
  for (int ds = 0; ds < 8; ++ds)
#pragma unroll
    for (int r = 0; r < 8; ++r)
      op[(size_t)(qb + r + 8 * hl) * DIM + ds * 16 + n] = oacc[ds][r];
}

extern "C" void kernel_launch(void* const* d_in, const int* in_sizes, int n_in,
                              void* d_out, int out_size, void* d_ws, size_t ws_size,
                              hipStream_t stream) {
  (void)in_sizes; (void)n_in; (void)out_size; (void)ws_size;
  const float* Q = (const float*)d_in[0];
  const float* K = (const float*)d_in[1];
  const float* V = (const float*)d_in[2];
  float* Out = (float*)d_out;

  float* Mw = (float*)d_ws;                 // [BH*SEQ] column maxes
  float* Di = Mw + (size_t)BH * SEQ;        // [BH*SEQ] reciprocal denominators

  dim3 block(256);                          // 8 wave32 per block
  dim3 grid((BH * T16) / 8);                // 4096 waves -> 512 blocks

  colsoftmax_stats_kernel<<<grid, block, 0, stream>>>(Q, K, Mw, Di);
  colsoftmax_av_kernel<<<grid, block, 0, stream>>>(Q, K, V, Mw, Di, Out);
}